// MultiBoxLoss_64407329571001
// MI455X (gfx1250) — compile-verified
//
#include <hip/hip_runtime.h>
#include <hip/hip_bf16.h>

#define BB 32
#define PP 131072
#define NBLK (512 * 32)            // pass A/C grid blocks: (PP/256) x BB

// ---------------- workspace layout ----------------
#define KEYS_BYTES   ((size_t)BB * PP * 4)            // 16 MiB of u32 keys
#define PART_SL1_OFF (KEYS_BYTES)                     // NBLK floats
#define PART_CE_OFF  (PART_SL1_OFF + (size_t)NBLK * 4)
#define CTRL_OFF     (PART_CE_OFF + (size_t)NBLK * 4)

struct Ctrl {
  int      num_pos[BB];
  unsigned thr_key[BB];
  unsigned r_rem[BB];
  unsigned tie_ticket[BB];
  unsigned sel_cnt;
  unsigned pad[3];
};

// ---------------- wave reduction via V_WMMA_F32_16X16X4_F32 ----------------
typedef float v2f __attribute__((ext_vector_type(2)));
typedef float v8f __attribute__((ext_vector_type(8)));

__device__ inline float waveReduceSum(float v) {
#if defined(__gfx1250__) && __has_builtin(__builtin_amdgcn_wmma_f32_16x16x4_f32)
  // A = ones(16x4), B slot0 = v, slot1 = 0  =>  D[m][n] = column sums of B.
  // D's first VGPR across 32 lanes covers each of the 16 columns exactly twice,
  // so sum over lanes of d[0] == 2 * sum(v) regardless of exact slot mapping.
  v2f a;  a[0] = 1.0f; a[1] = 1.0f;
  v2f bm; bm[0] = v;   bm[1] = 0.0f;
  v8f c = {0.f, 0.f, 0.f, 0.f, 0.f, 0.f, 0.f, 0.f};
  c = __builtin_amdgcn_wmma_f32_16x16x4_f32(false, a, false, bm, (short)0, c,
                                            false, false);
  float s = c[0];
#pragma unroll
  for (int off = 16; off; off >>= 1) s += __shfl_xor(s, off, 32);
  return 0.5f * s;
#else
  float s = v;
#pragma unroll
  for (int off = 16; off; off >>= 1) s += __shfl_xor(s, off, 32);
  return s;
#endif
}

// ---------------- init ----------------
__global__ void init_kernel(Ctrl* c) {
  int t = threadIdx.x;
  if (t < BB) { c->num_pos[t] = 0; c->tie_ticket[t] = 0; }
  if (t == 0) c->sel_cnt = 0;
}

// ---------------- pass A: stream everything once ----------------
__global__ __launch_bounds__(256) void pass_a(const float4* __restrict__ loc,
                                              const float4* __restrict__ loct,
                                              const float2* __restrict__ conf,
                                              const int*    __restrict__ ct,
                                              unsigned*     __restrict__ keys,
                                              Ctrl* __restrict__ ctrl,
                                              float* __restrict__ psl1) {
  const int b = blockIdx.y;
  const long long idx =
      (long long)b * PP + (long long)blockIdx.x * blockDim.x + threadIdx.x;

  const int t   = ct[idx];
  const int pos = (t > 0);

  // smooth L1 on positives
  float4 x = loc[idx], y = loct[idx];
  float d0 = fabsf(x.x - y.x), d1 = fabsf(x.y - y.y);
  float d2 = fabsf(x.z - y.z), d3 = fabsf(x.w - y.w);
  float e = (d0 < 1.f ? 0.5f * d0 * d0 : d0 - 0.5f) +
            (d1 < 1.f ? 0.5f * d1 * d1 : d1 - 0.5f) +
            (d2 < 1.f ? 0.5f * d2 * d2 : d2 - 0.5f) +
            (d3 < 1.f ? 0.5f * d3 * d3 : d3 - 0.5f);
  float s = pos ? e : 0.f;

  // score key: raw = logsumexp(conf) - conf[gt]  (>= 0); MSB = positive flag
  float2 c2 = conf[idx];
  float m   = fmaxf(c2.x, c2.y);
  float lse = m + logf(expf(c2.x - m) + expf(c2.y - m));
  float raw = lse - (pos ? c2.y : c2.x);
  keys[idx] = pos ? 0x80000000u : __float_as_uint(raw);

  // block reduction (deterministic slots, int atomics only)
  float wsum = waveReduceSum(s);
  int   wcnt = __popcll(__ballot(pos));
  __shared__ float sh_f[8];
  __shared__ int   sh_i[8];
  int lane = threadIdx.x & 31, wid = threadIdx.x >> 5;
  if (lane == 0) { sh_f[wid] = wsum; sh_i[wid] = wcnt; }
  __syncthreads();
  if (threadIdx.x == 0) {
    float tot = 0.f; int cnt = 0;
    for (int i = 0; i < 8; ++i) { tot += sh_f[i]; cnt += sh_i[i]; }
    psl1[blockIdx.y * gridDim.x + blockIdx.x] = tot;
    if (cnt) atomicAdd(&ctrl->num_pos[b], cnt);
  }
}

// ---------------- pass B: per-row radix select (4 x 8-bit rounds) ----------------
__global__ __launch_bounds__(1024) void pass_b(const unsigned* __restrict__ keys,
                                               Ctrl* __restrict__ ctrl) {
  const int b   = blockIdx.x;
  const int tid = threadIdx.x;
  __shared__ unsigned hist[256];
  __shared__ unsigned sh_prefix, sh_krem, sh_done;

  if (tid == 0) {
    long long k = 3LL * (long long)ctrl->num_pos[b];
    if (k > PP - 1) k = PP - 1;
    if (k <= 0) {
      ctrl->thr_key[b] = 0xFFFFFFFFu; ctrl->r_rem[b] = 0; sh_done = 1;
    } else {
      sh_krem = (unsigned)k; sh_prefix = 0; sh_done = 0;
    }
  }
  __syncthreads();
  if (sh_done) return;

  const unsigned* kk = keys + (long long)b * PP;
  for (int round = 0; round < 4; ++round) {
    const int shift = 24 - 8 * round;
    for (int i = tid; i < 256; i += 1024) hist[i] = 0;
    __syncthreads();
    const unsigned pref = sh_prefix;
    for (int i = tid; i < PP; i += 1024) {
      unsigned key = kk[i];
      unsigned ek  = (key & 0x80000000u) ? 0u : key;  // positives rank as 0
      unsigned hi  = (ek >> shift) >> 8;              // two legal shifts (round 0 -> 0)
      if (hi == pref) atomicAdd(&hist[(ek >> shift) & 0xFFu], 1u);
    }
    __syncthreads();
    if (tid == 0) {
      unsigned krem = sh_krem, cum = 0; int sel = 0;
      for (int v = 255; v >= 0; --v) {
        unsigned h = hist[v];
        if (cum + h >= krem) { sel = v; sh_krem = krem - cum; break; }
        cum += h;
      }
      sh_prefix = (sh_prefix << 8) | (unsigned)sel;
    }
    __syncthreads();
  }
  if (tid == 0) { ctrl->thr_key[b] = sh_prefix; ctrl->r_rem[b] = sh_krem; }
}

// ---------------- pass C: selection + CE accumulation ----------------
__global__ __launch_bounds__(256) void pass_c(const float2*  __restrict__ conf,
                                              const unsigned* __restrict__ keys,
                                              Ctrl* __restrict__ ctrl,
                                              float* __restrict__ pce) {
  const int b = blockIdx.y;
  const long long idx =
      (long long)b * PP + (long long)blockIdx.x * blockDim.x + threadIdx.x;

  const unsigned key = keys[idx];
  const int      pos = (int)(key >> 31);
  const unsigned ek  = pos ? 0u : key;
  const unsigned thr = ctrl->thr_key[b];
  const unsigned r   = ctrl->r_rem[b];

  int neg = 0;
  if (ek > thr) {
    neg = 1;
  } else if (ek == thr && r > 0) {
    // wave-aggregated tie ticketing: exactly r tied elements become neg.
    unsigned long long bm = __ballot(1);
    unsigned mask = (unsigned)bm;
    int lane = threadIdx.x & 31;
    int lead = __ffsll((long long)bm) - 1;
    int myrank = __popc(mask & ((lane == 31) ? 0x7FFFFFFFu : ((1u << lane) - 1u)));
    unsigned base = 0;
    if (lane == lead)
      base = atomicAdd(&ctrl->tie_ticket[b], (unsigned)__popc(mask));
    base = __shfl(base, lead, 32);
    neg = (base + (unsigned)myrank) < r;
  }
  const int sel = pos | neg;

  float2 c2 = conf[idx];
  float m   = fmaxf(c2.x, c2.y);
  float lse = m + logf(expf(c2.x - m) + expf(c2.y - m));
  float ce  = sel ? (lse - (pos ? c2.y : c2.x)) : 0.f;

  float wsum = waveReduceSum(ce);
  int   wcnt = __popcll(__ballot(sel));
  __shared__ float sh_f[8];
  __shared__ int   sh_i[8];
  int lane = threadIdx.x & 31, wid = threadIdx.x >> 5;
  if (lane == 0) { sh_f[wid] = wsum; sh_i[wid] = wcnt; }
  __syncthreads();
  if (threadIdx.x == 0) {
    float tot = 0.f; int cnt = 0;
    for (int i = 0; i < 8; ++i) { tot += sh_f[i]; cnt += sh_i[i]; }
    pce[blockIdx.y * gridDim.x + blockIdx.x] = tot;
    if (cnt) atomicAdd(&ctrl->sel_cnt, (unsigned)cnt);
  }
}

// ---------------- finalize: deterministic fixed-order grand totals ----------------
__global__ __launch_bounds__(256) void finalize_kernel(const float* __restrict__ psl1,
                                                       const float* __restrict__ pce,
                                                       const Ctrl* __restrict__ ctrl,
                                                       float* __restrict__ out) {
  __shared__ float sh1[8], sh2[8];
  int tid = threadIdx.x, lane = tid & 31, wid = tid >> 5;
  float s1 = 0.f, s2 = 0.f;
  for (int i = tid; i < NBLK; i += 256) { s1 += psl1[i]; s2 += pce[i]; }
  float w1 = waveReduceSum(s1);
  float w2 = waveReduceSum(s2);
  if (lane == 0) { sh1[wid] = w1; sh2[wid] = w2; }
  __syncthreads();
  if (tid == 0) {
    float t1 = 0.f, t2 = 0.f;
    for (int i = 0; i < 8; ++i) { t1 += sh1[i]; t2 += sh2[i]; }
    int np = 0;
    for (int b = 0; b < BB; ++b) np += ctrl->num_pos[b];
    float N = (float)np;
    out[0] = t1 / (N * 4.0f) / N;                       // loss_loc
    out[1] = t2 / (float)ctrl->sel_cnt / N;             // loss_conf
  }
}

extern "C" void kernel_launch(void* const* d_in, const int* in_sizes, int n_in,
                              void* d_out, int out_size, void* d_ws, size_t ws_size,
                              hipStream_t stream) {
  const float* loc  = (const float*)d_in[0];
  const float* conf = (const float*)d_in[1];
  const float* loct = (const float*)d_in[2];
  const int*   ct   = (const int*)d_in[3];
  float*       out  = (float*)d_out;

  char*     ws   = (char*)d_ws;
  unsigned* keys = (unsigned*)ws;
  float*    psl1 = (float*)(ws + PART_SL1_OFF);
  float*    pce  = (float*)(ws + PART_CE_OFF);
  Ctrl*     ctrl = (Ctrl*)(ws + CTRL_OFF);

  init_kernel<<<1, 64, 0, stream>>>(ctrl);
  pass_a<<<dim3(PP / 256, BB), 256, 0, stream>>>(
      (const float4*)loc, (const float4*)loct, (const float2*)conf, ct, keys,
      ctrl, psl1);
  pass_b<<<BB, 1024, 0, stream>>>(keys, ctrl);
  pass_c<<<dim3(PP / 256, BB), 256, 0, stream>>>((const float2*)conf, keys,
                                                 ctrl, pce);
  finalize_kernel<<<1, 256, 0, stream>>>(psl1, pce, ctrl, out);
}